// SimplifiedMambaBlock_63376537419871
// MI455X (gfx1250) — compile-verified
//
#include <hip/hip_runtime.h>
#include <hip/hip_bf16.h>

typedef float v2f __attribute__((ext_vector_type(2)));
typedef float v4f __attribute__((ext_vector_type(4)));
typedef float v8f __attribute__((ext_vector_type(8)));
typedef int   b128_t __attribute__((vector_size(16)));   // matches builtin param V4i

#define DIM     768
#define DSTATE  16
#define DCONV   4
#define E_DIM   1536
#define BATCH   2
#define SEQ     2048
#define NROW    (BATCH * SEQ)   /* 4096 rows of flattened (B,L) */
#define EPSF    1e-6f

#define KT      32              /* K-chunk staged in LDS        */
#define LDST    36              /* padded LDS row stride (floats): 16B-aligned
                                   rows, 36r mod 64 distinct for r=0..15 ->
                                   conflict-free ds_load_b64 fragment reads   */

#if defined(__HIP_DEVICE_COMPILE__) && __has_builtin(__builtin_amdgcn_global_load_async_to_lds_b128)
#define HAS_ASYNC_LDS 1
#else
#define HAS_ASYNC_LDS 0
#endif

// ---------------------------------------------------------------------------
// Async (or fallback) 16-byte global -> LDS copy, per lane.
// Builtin signature: (AS1 int4*, AS3 int4*, imm offset, imm cpol)
// ---------------------------------------------------------------------------
__device__ __forceinline__ void copy16_g2l(const float* g, float* l) {
#if HAS_ASYNC_LDS
  b128_t* gg = (b128_t*)const_cast<float*>(g);  // pointee bitcast (generic AS)
  b128_t* ll = (b128_t*)l;
  __builtin_amdgcn_global_load_async_to_lds_b128(
      (__attribute__((address_space(1))) b128_t*)gg,   // addrspacecast
      (__attribute__((address_space(3))) b128_t*)ll,   // addrspacecast
      0, 0);
#else
  *(v4f*)l = *(const v4f*)g;
#endif
}

__device__ __forceinline__ void wait_async_lds() {
#if HAS_ASYNC_LDS
#if __has_builtin(__builtin_amdgcn_s_wait_asynccnt)
  __builtin_amdgcn_s_wait_asynccnt(0);
#else
  asm volatile("s_wait_asynccnt 0x0" ::: "memory");
#endif
#endif
}

// ---------------------------------------------------------------------------
// Kernel 1: RMSNorm   xn = gamma * x / (rms + eps),  rms = sqrt(mean(x^2))
// ---------------------------------------------------------------------------
__global__ void rmsnorm_kernel(const float* __restrict__ x,
                               const float* __restrict__ gamma,
                               float* __restrict__ xn) {
  const int row = blockIdx.x;
  const float* xr   = x  + (size_t)row * DIM;
  float*       outr = xn + (size_t)row * DIM;

  __shared__ float red[256];
  float ss = 0.f;
  for (int i = threadIdx.x; i < DIM; i += 256) {
    float v = xr[i];
    ss += v * v;
  }
  red[threadIdx.x] = ss;
  __syncthreads();
  for (int s = 128; s > 0; s >>= 1) {
    if (threadIdx.x < s) red[threadIdx.x] += red[threadIdx.x + s];
    __syncthreads();
  }
  const float rms   = sqrtf(red[0] * (1.0f / (float)DIM));
  const float scale = 1.0f / (rms + EPSF);
  for (int i = threadIdx.x; i < DIM; i += 256) {
    outr[i] = gamma[i] * xr[i] * scale;
  }
}

// ---------------------------------------------------------------------------
// Stage one K-chunk (64 rows x 32 cols) of A and B tiles into LDS.
// 128 threads: thread t -> row = t>>1, colgroup = (t&1)*16; 4x b128 each.
// ---------------------------------------------------------------------------
__device__ __forceinline__ void stage_tiles(const float* __restrict__ gA,
                                            const float* __restrict__ gB,
                                            int K, int k0, int tid,
                                            float* __restrict__ lA,
                                            float* __restrict__ lB) {
  const int r  = tid >> 1;
  const int cg = (tid & 1) * 16;
  const float* ga = gA + (size_t)r * K + k0 + cg;
  const float* gb = gB + (size_t)r * K + k0 + cg;
  float* la = lA + r * LDST + cg;
  float* lb = lB + r * LDST + cg;
#pragma unroll
  for (int i = 0; i < 4; ++i) {
    copy16_g2l(ga + 4 * i, la + 4 * i);
    copy16_g2l(gb + 4 * i, lb + 4 * i);
  }
}

// ---------------------------------------------------------------------------
// Kernels 2 & 4: fp32 WMMA GEMM   C[M,N] = A[M,K] * W[N,K]^T + bias (+ resid)
// Block = 128 threads (4 waves) -> 64M x 64N tile. Wave w computes the
// 16x64 strip at rows [64*bx + 16w, +16). LDS double-buffered K-chunks,
// filled with async global->LDS b128 copies; fragments read via ds_load_b64.
//
// Fragment layouts (CDNA5 ISA 7.12.2, 32-bit, wave32):
//   A 16x4 : lane l -> M = l&15, holds K = k + 2*(l>>4) + {0,1}   (b64)
//   B 4x16 : lane l -> N = l&15, holds K = k + 2*(l>>4) + {0,1}   (b64)
//   C/D    : vgpr r, lane l -> M = r + 8*(l>>4), N = l&15
// ---------------------------------------------------------------------------
template <bool RESID>
__global__ __launch_bounds__(128)
void gemm_wmma_f32_kernel(const float* __restrict__ A,    // M x K
                          const float* __restrict__ W,    // N x K
                          const float* __restrict__ bias, // N
                          const float* __restrict__ R,    // M x N (or null)
                          float* __restrict__ Cout,       // M x N
                          int M, int N, int K) {
  __shared__ float lA[2][64 * LDST];
  __shared__ float lB[2][64 * LDST];

  const int tid   = threadIdx.x;
  const int lane  = tid & 31;
  const int wv    = tid >> 5;           // wave 0..3
  const int half  = lane >> 4;          // 0/1
  const int l16   = lane & 15;
  const int blockM = blockIdx.x * 64;
  const int tileN  = blockIdx.y * 64;
  const int tileM  = blockM + 16 * wv;

  const float* gA = A + (size_t)blockM * K;   // 64 A rows for this block
  const float* gB = W + (size_t)tileN * K;    // 64 W rows for this block

  v8f acc0 = {}, acc1 = {}, acc2 = {}, acc3 = {};

  // LDS read offsets (floats) for this lane's fragments
  const int aoff  = (16 * wv + l16) * LDST + 2 * half;
  const int boff0 = ( 0 + l16) * LDST + 2 * half;
  const int boff1 = (16 + l16) * LDST + 2 * half;
  const int boff2 = (32 + l16) * LDST + 2 * half;
  const int boff3 = (48 + l16) * LDST + 2 * half;

  // Prologue: stage chunk 0
  stage_tiles(gA, gB, K, 0, tid, lA[0], lB[0]);
  wait_async_lds();
  __syncthreads();

  const int nchunk = K / KT;
  for (int c = 0; c < nchunk; ++c) {
    const int cur = c & 1;
    // Prefetch next chunk into the other buffer (overlaps with compute)
    if (c + 1 < nchunk) {
      stage_tiles(gA, gB, K, (c + 1) * KT, tid, lA[cur ^ 1], lB[cur ^ 1]);
    }

    const float* la = lA[cur];
    const float* lb = lB[cur];
#pragma unroll
    for (int kk = 0; kk < KT; kk += 4) {
      v2f a  = *(const v2f*)(la + aoff  + kk);
      v2f b0 = *(const v2f*)(lb + boff0 + kk);
      v2f b1 = *(const v2f*)(lb + boff1 + kk);
      v2f b2 = *(const v2f*)(lb + boff2 + kk);
      v2f b3 = *(const v2f*)(lb + boff3 + kk);
      acc0 = __builtin_amdgcn_wmma_f32_16x16x4_f32(false, a, false, b0, (short)0, acc0, false, false);
      acc1 = __builtin_amdgcn_wmma_f32_16x16x4_f32(false, a, false, b1, (short)0, acc1, false, false);
      acc2 = __builtin_amdgcn_wmma_f32_16x16x4_f32(false, a, false, b2, (short)0, acc2, false, false);
      acc3 = __builtin_amdgcn_wmma_f32_16x16x4_f32(false, a, false, b3, (short)0, acc3, false, false);
    }

    wait_async_lds();   // next chunk's async copies have landed
    __syncthreads();    // all waves done reading cur / writing next
  }

  // Epilogue: D layout -> row = tileM + r + 8*half, col = tileN + l16 + 16*j
  const int c0 = tileN + l16;
  const float b0s = bias[c0 +  0];
  const float b1s = bias[c0 + 16];
  const float b2s = bias[c0 + 32];
  const float b3s = bias[c0 + 48];

#pragma unroll
  for (int r = 0; r < 8; ++r) {
    const int row   = tileM + r + 8 * half;
    const size_t ro = (size_t)row * N;
    float o0 = acc0[r] + b0s;
    float o1 = acc1[r] + b1s;
    float o2 = acc2[r] + b2s;
    float o3 = acc3[r] + b3s;
    if (RESID) {
      o0 += R[ro + c0 +  0];
      o1 += R[ro + c0 + 16];
      o2 += R[ro + c0 + 32];
      o3 += R[ro + c0 + 48];
    }
    Cout[ro + c0 +  0] = o0;
    Cout[ro + c0 + 16] = o1;
    Cout[ro + c0 + 32] = o2;
    Cout[ro + c0 + 48] = o3;
  }
}

// ---------------------------------------------------------------------------
// Kernel 3: fused causal depthwise conv1d (4-tap sliding register window)
// + sequential SSM scan, one thread per (b, e) channel, in-place xp -> y.
// ---------------------------------------------------------------------------
__device__ __forceinline__ float sigmoidf(float v) {
  return 1.0f / (1.0f + expf(-v));
}

__global__ void conv_ssm_kernel(float* __restrict__ xp,          // B x L x E
                                const float* __restrict__ convw, // E x 4
                                const float* __restrict__ Amat,  // E x 16
                                const float* __restrict__ Bmat,  // E x 16
                                const float* __restrict__ Cmat)  // E x 16
{
  const int g = blockIdx.x * blockDim.x + threadIdx.x;
  if (g >= BATCH * E_DIM) return;
  const int b = g / E_DIM;
  const int e = g % E_DIM;

  const float w0 = convw[e * DCONV + 0];
  const float w1 = convw[e * DCONV + 1];
  const float w2 = convw[e * DCONV + 2];
  const float w3 = convw[e * DCONV + 3];

  float sA[DSTATE], sB[DSTATE], sC[DSTATE], h[DSTATE];
#pragma unroll
  for (int n = 0; n < DSTATE; ++n) {
    sA[n] = sigmoidf(Amat[e * DSTATE + n]);
    sB[n] = sigmoidf(Bmat[e * DSTATE + n]);
    sC[n] = sigmoidf(Cmat[e * DSTATE + n]);
    h[n]  = 0.0f;
  }

  float p1 = 0.f, p2 = 0.f, p3 = 0.f;  // xp[t-1], xp[t-2], xp[t-3]
  float* col = xp + (size_t)b * SEQ * E_DIM + e;

  float cur = col[0];
  for (int t = 0; t < SEQ; ++t) {
    const float nxt = (t + 1 < SEQ) ? col[(size_t)(t + 1) * E_DIM] : 0.f;

    const float xc = fmaf(w0, p3, fmaf(w1, p2, fmaf(w2, p1, w3 * cur)));
    p3 = p2; p2 = p1; p1 = cur;

    float y = 0.f;
#pragma unroll
    for (int n = 0; n < DSTATE; ++n) {
      h[n] = fmaf(sA[n], h[n], sB[n] * xc);
      y    = fmaf(h[n], sC[n], y);
    }
    col[(size_t)t * E_DIM] = y;
    cur = nxt;
  }
}

// ---------------------------------------------------------------------------
// Host-side orchestration
// ---------------------------------------------------------------------------
extern "C" void kernel_launch(void* const* d_in, const int* in_sizes, int n_in,
                              void* d_out, int out_size, void* d_ws, size_t ws_size,
                              hipStream_t stream) {
  const float* x      = (const float*)d_in[0];  // (B, L, DIM)
  const float* gamma  = (const float*)d_in[1];  // (DIM,)
  const float* W_in   = (const float*)d_in[2];  // (E, DIM)
  const float* b_in   = (const float*)d_in[3];  // (E,)
  const float* conv_w = (const float*)d_in[4];  // (E, 4)
  const float* Amat   = (const float*)d_in[5];  // (E, 16)
  const float* Bmat   = (const float*)d_in[6];  // (E, 16)
  const float* Cmat   = (const float*)d_in[7];  // (E, 16)
  const float* W_out  = (const float*)d_in[8];  // (DIM, E)
  const float* b_out  = (const float*)d_in[9];  // (DIM,)
  float* out = (float*)d_out;                   // (B, L, DIM)

  float* xn = (float*)d_ws;                     // NROW x DIM
  float* xp = xn + (size_t)NROW * DIM;          // NROW x E (becomes y in place)

  // 1) RMSNorm
  rmsnorm_kernel<<<NROW, 256, 0, stream>>>(x, gamma, xn);

  // 2) xp = xn @ W_in^T + b_in     (M=4096, N=1536, K=768)
  {
    dim3 grid(NROW / 64, E_DIM / 64);
    gemm_wmma_f32_kernel<false><<<grid, 128, 0, stream>>>(
        xn, W_in, b_in, nullptr, xp, NROW, E_DIM, DIM);
  }

  // 3) causal conv + SSM scan (in place: xp -> y)
  {
    const int nch = BATCH * E_DIM;
    conv_ssm_kernel<<<(nch + 255) / 256, 256, 0, stream>>>(
        xp, conv_w, Amat, Bmat, Cmat);
  }

  // 4) out = y @ W_out^T + b_out + x   (M=4096, N=768, K=1536)
  {
    dim3 grid(NROW / 64, DIM / 64);
    gemm_wmma_f32_kernel<true><<<grid, 128, 0, stream>>>(
        xp, W_out, b_out, x, out, NROW, DIM, E_DIM);
  }
}